// ZloebuchyAdapter_26594437497233
// MI455X (gfx1250) — compile-verified
//
#include <hip/hip_runtime.h>

typedef __attribute__((ext_vector_type(2))) float v2f;
typedef __attribute__((ext_vector_type(8))) float v8f;

#define MAT_N 4096
#define NBLK 256
#define SPLIT 4
#define FLAT_STRIDE 20          // 16x16 tile-chunk staging, padded
#define HID_STRIDE 100          // hid[t][h] staging, padded (conflict-free mod 64 banks)
#define WAVE_LDS (16*FLAT_STRIDE + 16*HID_STRIDE)   // 320 + 1600 = 1920 floats

__global__ __launch_bounds__(256, 1)
void hyper_adapter_kernel(const float* __restrict__ sm,
                          const float* __restrict__ w1,
                          const float* __restrict__ b1,
                          const float* __restrict__ emb_h,
                          const float* __restrict__ emb_w,
                          const float* __restrict__ w2,
                          const float* __restrict__ b2,
                          const float* __restrict__ scales,
                          float* __restrict__ out)
{
    __shared__ float lds[8 * WAVE_LDS];
    const int tid  = threadIdx.x;
    const int wave = tid >> 5;
    const int lane = tid & 31;
    const int m    = lane & 15;     // N index / tile index within half
    const int hs   = lane >> 4;     // half-select: 0 = lanes 0-15, 1 = lanes 16-31
    const bool isH = (m < 8);

    float* flatbuf = lds + wave * WAVE_LDS;        // [16][FLAT_STRIDE]
    float* hidb    = flatbuf + 16 * FLAT_STRIDE;   // [16][HID_STRIDE] row-major hid[t][h]

    // per-lane base pointers: all loop offsets become DS-immediate constants
    float*       flat_st  = flatbuf + m * FLAT_STRIDE + 2 * hs;      // + 4j
    float*       hid_st   = hidb + 8 * hs * HID_STRIDE + m;          // + v*HID_STRIDE + 16n
    const float* hid_ld   = hidb + m * HID_STRIDE + 2 * hs;          // + 4j (contiguous b64 pair)
    float*       flat_rw  = flatbuf + 8 * hs * FLAT_STRIDE + m;      // + v*FLAT_STRIDE

    const int g = blockIdx.x;
    const int b = g >> 2;           // block 0..255
    const int quarter = g & 3;
    const int obi = b >> 4;
    const int ibi = b & 15;

    // ---- preload hypernet weights as WMMA B-fragments (block-constant) ----
    v2f B1[6][4];
    {
        const float* w1b = w1 + b * (16 * 96);
        #pragma unroll
        for (int n = 0; n < 6; ++n)
            #pragma unroll
            for (int j = 0; j < 4; ++j) {
                const int kk = 4*j + 2*hs;
                B1[n][j].x = w1b[kk * 96 + 16*n + m];
                B1[n][j].y = w1b[(kk + 1) * 96 + 16*n + m];
            }
    }
    v2f B2[24];
    {
        const float* w2b = w2 + b * (96 * 16);
        #pragma unroll
        for (int j = 0; j < 24; ++j) {
            const int kk = 4*j + 2*hs;
            B2[j].x = w2b[kk * 16 + m];
            B2[j].y = w2b[(kk + 1) * 16 + m];
        }
    }
    float b1v[6];
    #pragma unroll
    for (int n = 0; n < 6; ++n) b1v[n] = b1[b*96 + 16*n + m];
    const float b2v = b2[b*16 + m];
    const float scv = scales[b*16 + m];

    const int ch0 = quarter*64 + wave*8;   // first chunk for this wave
    const int m7  = m & 7;                 // always-in-bounds embedding column

    // software pipeline: A1 fragments for the first chunk
    v2f A1[4];
    {
        const int th = ch0 >> 2, tw0 = (ch0 & 3) * 16;
        #pragma unroll
        for (int j = 0; j < 4; ++j) {
            const int row = obi*256 + th*4 + j;
            const int col = ibi*256 + (tw0 + m)*4 + 2*hs;
            A1[j] = *(const v2f*)(sm + (size_t)row * MAT_N + col);
        }
    }

    #pragma unroll 1
    for (int c8 = 0; c8 < 8; ++c8) {
        const int ch  = ch0 + c8;                   // 0..255
        const int th  = ch >> 2;                    // tile-row in block (shared by chunk)
        const int tw0 = (ch & 3) * 16;              // first tile-col of chunk

        // branchless positional-embedding gather (in flight during GEMM1):
        // lanes m<8 use emb_h[b][th][m]; lanes m>=8 use emb_w[b][tw0+v+8hs][m-8].
        // Both loaded unpredicated at column m&7 (always valid), then selected.
        const float ehv = emb_h[((size_t)b*64 + th)*8 + m7];
        const float* ewp = emb_w + ((size_t)b*64 + tw0 + 8*hs)*8 + m7;
        float embv[8];
        #pragma unroll
        for (int v = 0; v < 8; ++v) {
            const float ewv = ewp[v * 8];           // constant immediate offsets
            embv[v] = isH ? ehv : ewv;              // v_cndmask, no exec masking
        }

        // ---- stage flat chunk into LDS (residual source) ----
        #pragma unroll
        for (int j = 0; j < 4; ++j)
            *(v2f*)(flat_st + 4*j) = A1[j];

        // ---- GEMM1: hid(16x96) = flat(16x16) x w1(16x96), 24 WMMAs ----
        v8f C[6];
        #pragma unroll
        for (int n = 0; n < 6; ++n) {
            v8f acc = {};
            #pragma unroll
            for (int j = 0; j < 4; ++j)
                acc = __builtin_amdgcn_wmma_f32_16x16x4_f32(
                          false, A1[j], false, B1[n][j], (short)0, acc, false, false);
            C[n] = acc;
        }

        // ---- prefetch next chunk's A1 (hides HBM latency behind GEMM2) ----
        {
            const int chn = (c8 < 7) ? (ch + 1) : ch;
            const int thn = chn >> 2, tw0n = (chn & 3) * 16;
            #pragma unroll
            for (int j = 0; j < 4; ++j) {
                const int row = obi*256 + thn*4 + j;
                const int col = ibi*256 + (tw0n + m)*4 + 2*hs;
                A1[j] = *(const v2f*)(sm + (size_t)row * MAT_N + col);
            }
        }

        // ---- epilogue 1: bias + emb + ELU (branchless), stage hid[t][h] ----
        // elu(x) = max(x,0) + min(exp(x),1) - 1   (exact for both signs)
        #pragma unroll
        for (int n = 0; n < 6; ++n) {
            #pragma unroll
            for (int v = 0; v < 8; ++v) {
                float x = C[n][v] + b1v[n];
                if (n == 0) x += embv[v];
                const float e = __expf(x);                      // v_exp_f32 (TRANS)
                const float r = fmaxf(x, 0.0f) + (fminf(e, 1.0f) - 1.0f);
                // hid[t][h] with t = v + 8*hs, h = 16n + m
                hid_st[v * HID_STRIDE + 16*n] = r;
            }
        }

        asm volatile("s_wait_dscnt 0" ::: "memory");   // cross-lane LDS handoff (in-order DS)

        // ---- GEMM2: out(16x16) = hid(16x96) x w2(96x16), 24 WMMAs ----
        v8f D = {};
        #pragma unroll
        for (int j = 0; j < 24; ++j) {
            const v2f a = *(const v2f*)(hid_ld + 4*j);   // single ds_load_b64, aligned pair
            D = __builtin_amdgcn_wmma_f32_16x16x4_f32(
                    false, a, false, B2[j], (short)0, D, false, false);
        }

        // ---- epilogue 2: bias + scaled residual; overwrite flatbuf with result ----
        #pragma unroll
        for (int v = 0; v < 8; ++v) {
            const float res = flat_rw[v * FLAT_STRIDE];
            const float val = __builtin_fmaf(res, scv, D[v] + b2v);
            flat_rw[v * FLAT_STRIDE] = val;
        }

        asm volatile("s_wait_dscnt 0" ::: "memory");

        // ---- coalesced untiled store: chunk = 4 global rows x 256B ----
        #pragma unroll
        for (int p = 0; p < 2; ++p) {
            const int f = lane + 32*p;     // float4 index 0..63
            const int r = f >> 4;          // row within tile 0..3
            const int q = f & 15;          // tile within chunk
            const float4 val = *(const float4*)(flatbuf + q*FLAT_STRIDE + r*4);
            const int row = obi*256 + th*4 + r;
            const int col = ibi*256 + (tw0 + q)*4;
            *(float4*)(out + (size_t)row * MAT_N + col) = val;
        }
        asm volatile("s_wait_dscnt 0" ::: "memory");
    }
}

extern "C" void kernel_launch(void* const* d_in, const int* in_sizes, int n_in,
                              void* d_out, int out_size, void* d_ws, size_t ws_size,
                              hipStream_t stream) {
    const float* sm = (const float*)d_in[0];
    const float* w1 = (const float*)d_in[1];
    const float* b1 = (const float*)d_in[2];
    const float* eh = (const float*)d_in[3];
    const float* ew = (const float*)d_in[4];
    const float* w2 = (const float*)d_in[5];
    const float* b2 = (const float*)d_in[6];
    const float* sc = (const float*)d_in[7];
    float* out = (float*)d_out;
    hipLaunchKernelGGL(hyper_adapter_kernel, dim3(NBLK * SPLIT), dim3(256), 0, stream,
                       sm, w1, b1, eh, ew, w2, b2, sc, out);
}